// HGNNPConv_69123203662122
// MI455X (gfx1250) — compile-verified
//
#include <hip/hip_runtime.h>
#include <hip/hip_bf16.h>

// ---------------------------------------------------------------------------
// HGNNP conv:  H = X@W + b ;  e = mean_scatter(H[v_idx], e_idx) ;
//              v = mean_scatter(e[e_idx], v_idx) ;  out = leaky_relu(v, 0.01)
// Dims: N_V=50000, N_E=10000, N_P=800000, D=256 (all fp32).
// ---------------------------------------------------------------------------

typedef float v2f __attribute__((ext_vector_type(2)));
typedef float v8f __attribute__((ext_vector_type(8)));

#define HG_D      256
#define HG_NE     10000
#define HG_SLOPE  0.01f

// Guaranteed-native fp32 global atomic add (coarse-grained device memory only).
__device__ __forceinline__ void hg_atomic_add(float* p, float v) {
#if defined(__HIP_DEVICE_COMPILE__)
  unsafeAtomicAdd(p, v);     // lowers to global_atomic_add_f32
#else
  atomicAdd(p, v);
#endif
}

// ---------------------------------------------------------------------------
// GEMM: one wave computes a 16x16 tile of H = X*W + b using
// V_WMMA_F32_16X16X4_F32 (exact fp32, K-loop of 64 steps).
//
// A (16x4 f32) lane layout (ISA 7.12.2): lanes 0-15 -> {K=0,K=1} in v0,v1 for
// row M=lane; lanes 16-31 -> {K=2,K=3} for row M=lane-16.
// B (4x16 f32): lanes 0-15 -> rows K=0,1 at col N=lane; lanes 16-31 -> K=2,3.
// C/D (16x16 f32, 8 VGPRs): VGPR i -> (M = 8*(lane>=16) + i, N = lane&15).
// ---------------------------------------------------------------------------
__global__ __launch_bounds__(32)
void hg_gemm_wmma(const float* __restrict__ X, const float* __restrict__ W,
                  const float* __restrict__ b, float* __restrict__ H) {
  const int lane = threadIdx.x;        // 0..31
  const int half = lane >> 4;          // 0 or 1
  const int l15  = lane & 15;
  const int m0   = blockIdx.x * 16;
  const int n0   = blockIdx.y * 16;

  const int arow = m0 + l15;           // A-matrix row for this lane
  const int col  = n0 + l15;           // B/C/D column for this lane

  // D = A*B + C, so seed C with the bias (same value down the column).
  const float bias = b[col];
  v8f acc;
#pragma unroll
  for (int i = 0; i < 8; ++i) acc[i] = bias;

  const float* Arow = X + (size_t)arow * HG_D;

  for (int k = 0; k < HG_D / 4; ++k) {
    const int kk = k * 4 + half * 2;               // this lane's two K slices
    v2f a = *(const v2f*)(Arow + kk);              // global_load_b64
    v2f bb;
    bb.x = W[(size_t)kk * HG_D + col];
    bb.y = W[(size_t)(kk + 1) * HG_D + col];
    acc = __builtin_amdgcn_wmma_f32_16x16x4_f32(
        /*neg_a=*/false, a, /*neg_b=*/false, bb,
        /*c_mod=*/(short)0, acc, /*reuse_a=*/false, /*reuse_b=*/false);
  }

  // Store: VGPR i holds (row m0 + half*8 + i, col).
  float* Hp = H + (size_t)(m0 + half * 8) * HG_D + col;
#pragma unroll
  for (int i = 0; i < 8; ++i) Hp[(size_t)i * HG_D] = acc[i];
}

// ---------------------------------------------------------------------------
// Scatter-add one 256-float row per incidence pair: one wave per pair,
// each lane owns 8 contiguous floats (two float4 loads, 8 f32 atomics).
// Reused for both directions (src/dst index arrays swapped).
// ---------------------------------------------------------------------------
__global__ __launch_bounds__(256)
void hg_scatter_add(const float* __restrict__ src, const int* __restrict__ src_idx,
                    const int* __restrict__ dst_idx, float* __restrict__ dst_acc,
                    float* __restrict__ dst_cnt, int n_pairs) {
  const int wave = blockIdx.x * (blockDim.x >> 5) + (threadIdx.x >> 5);
  const int lane = threadIdx.x & 31;
  if (wave >= n_pairs) return;

  const int s = src_idx[wave];
  const int d = dst_idx[wave];

  const float* sp = src + (size_t)s * HG_D + lane * 8;
  float*       dp = dst_acc + (size_t)d * HG_D + lane * 8;

  const float4 a = ((const float4*)sp)[0];
  const float4 c = ((const float4*)sp)[1];

  hg_atomic_add(dp + 0, a.x);
  hg_atomic_add(dp + 1, a.y);
  hg_atomic_add(dp + 2, a.z);
  hg_atomic_add(dp + 3, a.w);
  hg_atomic_add(dp + 4, c.x);
  hg_atomic_add(dp + 5, c.y);
  hg_atomic_add(dp + 6, c.z);
  hg_atomic_add(dp + 7, c.w);

  if (lane == 0) hg_atomic_add(dst_cnt + d, 1.0f);
}

// acc[i] /= max(cnt[i/256], 1)  (in place; used for the hyperedge mean)
__global__ __launch_bounds__(256)
void hg_seg_div(float* __restrict__ acc, const float* __restrict__ cnt, int n) {
  const int i = blockIdx.x * blockDim.x + threadIdx.x;
  if (i >= n) return;
  const float c = fmaxf(cnt[i >> 8], 1.0f);
  acc[i] = acc[i] / c;
}

// out[i] = leaky_relu(out[i] / max(cnt[i/256], 1))
__global__ __launch_bounds__(256)
void hg_finalize(float* __restrict__ out, const float* __restrict__ cnt, int n) {
  const int i = blockIdx.x * blockDim.x + threadIdx.x;
  if (i >= n) return;
  const float v = out[i] / fmaxf(cnt[i >> 8], 1.0f);
  out[i] = (v >= 0.0f) ? v : HG_SLOPE * v;
}

__global__ __launch_bounds__(256)
void hg_zero(float* __restrict__ p, long n) {
  long i = (long)blockIdx.x * blockDim.x + threadIdx.x;
  const long stride = (long)gridDim.x * blockDim.x;
  for (; i < n; i += stride) p[i] = 0.0f;
}

// ---------------------------------------------------------------------------
extern "C" void kernel_launch(void* const* d_in, const int* in_sizes, int n_in,
                              void* d_out, int out_size, void* d_ws, size_t ws_size,
                              hipStream_t stream) {
  const float* X     = (const float*)d_in[0];   // [N_V, 256]
  const float* W     = (const float*)d_in[1];   // [256, 256]
  const float* b     = (const float*)d_in[2];   // [256]
  const int*   v_idx = (const int*)d_in[3];     // [N_P]
  const int*   e_idx = (const int*)d_in[4];     // [N_P]

  const int N_V = in_sizes[0] / HG_D;           // 50000
  const int N_P = in_sizes[3];                  // 800000
  const int N_E = HG_NE;                        // 10000 (python scalar input)

  float* out = (float*)d_out;                   // doubles as vertex accumulator

  // Workspace layout (fp32): H | e_acc | e_cnt | v_cnt
  float* H     = (float*)d_ws;                       // N_V*256
  float* e_acc = H     + (size_t)N_V * HG_D;         // N_E*256
  float* e_cnt = e_acc + (size_t)N_E * HG_D;         // N_E
  float* v_cnt = e_cnt + (size_t)N_E;                // N_V

  // 1) zero the accumulators (must happen every call: graph replays)
  const long zcount = (long)N_E * HG_D + N_E + N_V;
  hg_zero<<<2048, 256, 0, stream>>>(e_acc, zcount);
  hg_zero<<<2048, 256, 0, stream>>>(out, (long)N_V * HG_D);

  // 2) H = X @ W + b  (WMMA fp32, one wave per 16x16 tile)
  dim3 ggrid(N_V / 16, HG_D / 16);
  hg_gemm_wmma<<<ggrid, 32, 0, stream>>>(X, W, b, H);

  // 3) v2e scatter-sum + counts
  const int waves_per_block = 8;
  const int sblocks = (N_P + waves_per_block - 1) / waves_per_block;
  hg_scatter_add<<<sblocks, 256, 0, stream>>>(H, v_idx, e_idx, e_acc, e_cnt, N_P);

  // 4) e_feat = e_acc / max(cnt,1)
  hg_seg_div<<<(N_E * HG_D + 255) / 256, 256, 0, stream>>>(e_acc, e_cnt, N_E * HG_D);

  // 5) e2v scatter-sum + counts (into d_out)
  hg_scatter_add<<<sblocks, 256, 0, stream>>>(e_acc, e_idx, v_idx, out, v_cnt, N_P);

  // 6) vertex mean + leaky relu (in place)
  hg_finalize<<<((long)N_V * HG_D + 255) / 256, 256, 0, stream>>>(out, v_cnt, N_V * HG_D);
}